// GemNet_5669356834240
// MI455X (gfx1250) — compile-verified
//
#include <hip/hip_runtime.h>

// GemNet triplet-interaction + atom-update, fp32 WMMA (16x16x4) on gfx1250.
// Memory-bound (~1.3 GB moved vs 34 GFLOP at 23.3 TB/s -> ~56us floor), so we
// keep fp32 precision and optimize data motion: m/h A-fragments hoisted so each
// big stream is read exactly once, L2-resident gather/scatter tables (x_down
// 38 MB, x_e 38 MB, h 10 MB all fit in 192 MB L2), streamed cbf/m with
// global_prefetch, and fused kernels so no E x 128 intermediate hits memory.

#define D  128
#define DT 16
#define NR 16
#define NS 16

typedef float v2f __attribute__((ext_vector_type(2)));
typedef float v8f __attribute__((ext_vector_type(8)));

#define WMMA_F32(a, b, c) \
  __builtin_amdgcn_wmma_f32_16x16x4_f32(false, (a), false, (b), (short)0, (c), false, false)

__device__ __forceinline__ float ssilu_f(float x) {
  // silu(x) / 0.6
  return (x * (1.0f / 0.6f)) / (1.0f + __expf(-x));
}

// A-fragment (16x4 f32): lane&15 = row, lane>>4 picks K pair {0,1} vs {2,3}
__device__ __forceinline__ v2f load_a(const float* __restrict__ src, int row0, int ld,
                                      int k0, int lane) {
  int r = lane & 15, hh = lane >> 4;
  const float* p = src + (size_t)(row0 + r) * ld + k0 + 2 * hh;
  return *(const v2f*)p;  // 8B aligned: ld and k0 even
}

// B-fragment (4x16 f32) from row-major W[K][ldn]: lane&15 = col
__device__ __forceinline__ v2f load_b(const float* __restrict__ W, int n0, int ldn,
                                      int k0, int lane) {
  int c = lane & 15, hh = lane >> 4;
  int k = k0 + 2 * hh;
  v2f b = { W[(size_t)k * ldn + n0 + c], W[(size_t)(k + 1) * ldn + n0 + c] };
  return b;
}

__global__ void k0_zero(float* __restrict__ p, size_t n) {
  size_t i = (size_t)blockIdx.x * blockDim.x + threadIdx.x;
  size_t stride = (size_t)gridDim.x * blockDim.x;
  for (; i < n; i += stride) p[i] = 0.0f;
}

// K1: x_down[E,16] = (ssilu(m@W_ba) * (rbf@W_rbf)) @ W_down, fused per 16-row tile.
__global__ void __launch_bounds__(128)
k1_edge_down(const float* __restrict__ m, const float* __restrict__ rbf,
             const float* __restrict__ W_ba, const float* __restrict__ W_rbf,
             const float* __restrict__ W_down, float* __restrict__ x_down, int E) {
  __shared__ float stage[4][16 * 16];  // per-wave C->A relayout buffer
  int lane = threadIdx.x & 31;
  int wave = threadIdx.x >> 5;
  int nt = E >> 4;
  int wid = blockIdx.x * 4 + wave;
  int nw  = gridDim.x * 4;
  float* st = stage[wave];

  for (int tile = wid; tile < nt; tile += nw) {
    int e0 = tile << 4;
    // Prefetch next tile's m rows (stride known per wave).
    if (tile + nw < nt)
      __builtin_prefetch(m + (size_t)(e0 + (nw << 4)) * D, 0, 0);

    // Hoist ALL A-fragments of this m tile: read m exactly once (64 VGPRs).
    v2f am[32];
#pragma unroll
    for (int kc = 0; kc < 32; kc++) am[kc] = load_a(m, e0, D, 4 * kc, lane);
    v2f arb[4];
#pragma unroll
    for (int kc = 0; kc < 4; kc++) arb[kc] = load_a(rbf, e0, NR, 4 * kc, lane);

    v8f dacc = {};  // x_down 16x16 accumulator
#pragma unroll 1
    for (int n = 0; n < 8; n++) {
      int n0 = n * 16;
      // x_ba col-tile: m(16x128) @ W_ba[:, n0:n0+16]   (W is L0/L2-hot)
      v8f acc = {};
#pragma unroll
      for (int kc = 0; kc < 32; kc++) {
        v2f b = load_b(W_ba, n0, D, 4 * kc, lane);
        acc = WMMA_F32(am[kc], b, acc);
      }
      // rbf projection col-tile: rbf(16x16) @ W_rbf[:, n0:n0+16]
      v8f racc = {};
#pragma unroll
      for (int kc = 0; kc < 4; kc++) {
        v2f b = load_b(W_rbf, n0, D, 4 * kc, lane);
        racc = WMMA_F32(arb[kc], b, racc);
      }
      // elementwise ssilu*rbf, stage row-major into LDS (same-wave DS is in-order)
#pragma unroll
      for (int v = 0; v < 8; v++) {
        int row = v + 8 * (lane >> 4);
        int col = lane & 15;
        st[row * 16 + col] = ssilu_f(acc[v]) * racc[v];
      }
      // x_down += xba_tile(16x16) @ W_down[n0:n0+16, :]
#pragma unroll
      for (int kc = 0; kc < 4; kc++) {
        int r = lane & 15, hh = lane >> 4;
        v2f a = *(const v2f*)(st + r * 16 + 4 * kc + 2 * hh);
        v2f b = load_b(W_down, 0, DT, n0 + 4 * kc, lane);
        dacc = WMMA_F32(a, b, dacc);
      }
    }
#pragma unroll
    for (int v = 0; v < 8; v++) {
      int row = v + 8 * (lane >> 4), col = lane & 15;
      x_down[(size_t)(e0 + row) * DT + col] = dacc[v];
    }
  }
}

// K2: x_t = x_down[id3_ba] * (cbf@W_cbf); x_e = segment_sum(x_t, id3_ca) via atomics.
__global__ void __launch_bounds__(256)
k2_triplet(const float* __restrict__ cbf, const int* __restrict__ id3_ba,
           const int* __restrict__ id3_ca, const float* __restrict__ W_cbf,
           const float* __restrict__ x_down, float* __restrict__ x_e, int T) {
  int lane = threadIdx.x & 31;
  int wid = blockIdx.x * (blockDim.x >> 5) + (threadIdx.x >> 5);
  int nw  = gridDim.x * (blockDim.x >> 5);
  int nt  = T >> 4;

  v2f bw[4];  // W_cbf B-fragments, reused for all tiles
#pragma unroll
  for (int kc = 0; kc < 4; kc++) bw[kc] = load_b(W_cbf, 0, DT, 4 * kc, lane);

  for (int tile = wid; tile < nt; tile += nw) {
    int t0 = tile << 4;
    if (t0 + 272 <= T) __builtin_prefetch(cbf + (size_t)(t0 + 256) * NS, 0, 0);
    int myid_b = id3_ba[t0 + (lane & 15)];
    int myid_c = id3_ca[t0 + (lane & 15)];

    v8f c = {};
#pragma unroll
    for (int kc = 0; kc < 4; kc++) {
      v2f a = load_a(cbf, t0, NS, 4 * kc, lane);
      c = WMMA_F32(a, bw[kc], c);
    }
    int col = lane & 15;
#pragma unroll
    for (int v = 0; v < 8; v++) {
      int row = v + 8 * (lane >> 4);
      int eb = __shfl(myid_b, row, 32);
      int ec = __shfl(myid_c, row, 32);
      float g = x_down[(size_t)eb * DT + col];           // L2-resident gather (38 MB)
      atomicAdd(&x_e[(size_t)ec * DT + col], c[v] * g);  // L2-resident scatter
    }
  }
}

// K3: m_new = m + ssilu(x_e@W_up); h += scatter(m_new * (rbf@W_rbf_atom), idx_t)
__global__ void __launch_bounds__(256)
k3_edge_atom(const float* __restrict__ m, const float* __restrict__ rbf,
             const float* __restrict__ x_e, const float* __restrict__ W_up,
             const float* __restrict__ W_rbf_atom, const int* __restrict__ idx_t,
             float* __restrict__ h, int E) {
  int lane = threadIdx.x & 31;
  int wid = blockIdx.x * (blockDim.x >> 5) + (threadIdx.x >> 5);
  int nw  = gridDim.x * (blockDim.x >> 5);
  int nt  = E >> 4;

  for (int tile = wid; tile < nt; tile += nw) {
    int e0 = tile << 4;
    if (tile + nw < nt)
      __builtin_prefetch(m + (size_t)(e0 + (nw << 4)) * D, 0, 0);
    int myidx = idx_t[e0 + (lane & 15)];
    v2f axe[4], arb[4];
#pragma unroll
    for (int kc = 0; kc < 4; kc++) {
      axe[kc] = load_a(x_e, e0, DT, 4 * kc, lane);
      arb[kc] = load_a(rbf, e0, NR, 4 * kc, lane);
    }
    int col = lane & 15;
#pragma unroll 1
    for (int n = 0; n < 8; n++) {
      int n0 = n * 16;
      v8f up = {}, ra = {};
#pragma unroll
      for (int kc = 0; kc < 4; kc++) {
        up = WMMA_F32(axe[kc], load_b(W_up, n0, D, 4 * kc, lane), up);
        ra = WMMA_F32(arb[kc], load_b(W_rbf_atom, n0, D, 4 * kc, lane), ra);
      }
#pragma unroll
      for (int v = 0; v < 8; v++) {
        int row = v + 8 * (lane >> 4);
        float mv = m[(size_t)(e0 + row) * D + n0 + col];
        float mn = mv + ssilu_f(up[v]);
        int a = __shfl(myidx, row, 32);
        atomicAdd(&h[(size_t)a * D + n0 + col], mn * ra[v]);  // h is 10 MB, L2-hot
      }
    }
  }
}

// K4: out = ssilu(h @ W_atom)
__global__ void __launch_bounds__(256)
k4_atom_out(const float* __restrict__ h, const float* __restrict__ W_atom,
            float* __restrict__ out, int N) {
  int lane = threadIdx.x & 31;
  int wid = blockIdx.x * (blockDim.x >> 5) + (threadIdx.x >> 5);
  int nw  = gridDim.x * (blockDim.x >> 5);
  int nt  = N >> 4;

  for (int tile = wid; tile < nt; tile += nw) {
    int r0 = tile << 4;
    int col = lane & 15;

    // Hoist h A-fragments: read h tile once, reuse across 8 column tiles.
    v2f ah[32];
#pragma unroll
    for (int kc = 0; kc < 32; kc++) ah[kc] = load_a(h, r0, D, 4 * kc, lane);

#pragma unroll 1
    for (int n = 0; n < 8; n++) {
      int n0 = n * 16;
      v8f acc = {};
#pragma unroll
      for (int kc = 0; kc < 32; kc++) {
        v2f b = load_b(W_atom, n0, D, 4 * kc, lane);
        acc = WMMA_F32(ah[kc], b, acc);
      }
#pragma unroll
      for (int v = 0; v < 8; v++) {
        int row = v + 8 * (lane >> 4);
        out[(size_t)(r0 + row) * D + n0 + col] = ssilu_f(acc[v]);
      }
    }
  }
}

extern "C" void kernel_launch(void* const* d_in, const int* in_sizes, int n_in,
                              void* d_out, int out_size, void* d_ws, size_t ws_size,
                              hipStream_t stream) {
  const float* m          = (const float*)d_in[0];
  const float* rbf        = (const float*)d_in[1];
  const float* cbf        = (const float*)d_in[2];
  const int*   id3_ba     = (const int*)d_in[3];
  const int*   id3_ca     = (const int*)d_in[4];
  const int*   idx_t      = (const int*)d_in[5];
  const float* W_ba       = (const float*)d_in[6];
  const float* W_rbf      = (const float*)d_in[7];
  const float* W_down     = (const float*)d_in[8];
  const float* W_cbf      = (const float*)d_in[9];
  const float* W_up       = (const float*)d_in[10];
  const float* W_rbf_atom = (const float*)d_in[11];
  const float* W_atom     = (const float*)d_in[12];
  float* out = (float*)d_out;

  int E = in_sizes[0] / D;    // 600000
  int T = in_sizes[2] / NS;   // 8000000
  int N = out_size / D;       // 20000

  float* x_down = (float*)d_ws;                  // [E,16]
  float* x_e    = x_down + (size_t)E * DT;       // [E,16]
  float* h      = x_e + (size_t)E * DT;          // [N,128]

  size_t zn = (size_t)E * DT + (size_t)N * D;    // x_e and h are contiguous
  k0_zero<<<2048, 256, 0, stream>>>(x_e, zn);
  k1_edge_down<<<2048, 128, 0, stream>>>(m, rbf, W_ba, W_rbf, W_down, x_down, E);
  k2_triplet<<<4096, 256, 0, stream>>>(cbf, id3_ba, id3_ca, W_cbf, x_down, x_e, T);
  k3_edge_atom<<<2048, 256, 0, stream>>>(m, rbf, x_e, W_up, W_rbf_atom, idx_t, h, E);
  k4_atom_out<<<160, 256, 0, stream>>>(h, W_atom, out, N);
}